// MSSP_16887811407926
// MI455X (gfx1250) — compile-verified
//
#include <hip/hip_runtime.h>
#include <hip/hip_bf16.h>
#include <math.h>

typedef __attribute__((ext_vector_type(2))) float v2f;
typedef __attribute__((ext_vector_type(8))) float v8f;
typedef __attribute__((ext_vector_type(4))) int   v4i;

typedef __attribute__((address_space(1))) v4i* gv4i_p;   // global b128
typedef __attribute__((address_space(3))) v4i* sv4i_p;   // LDS b128

#define NB 64      // batch
#define SS 512     // sequence
#define HH 768     // hidden
#define PP 256     // pieces
#define VV 2187    // sememe classes

constexpr int KC     = 32;       // K chunk staged in LDS
constexpr int LDA    = 36;       // padded row stride: 144B, 16B-aligned, conflict-free
constexpr int NCHUNK = HH / KC;  // 24

// --- CDNA5 async global->LDS copy (GLOBAL_LOAD_ASYNC_TO_LDS_B128, ASYNCcnt) ---
// imm offset must be a frontend constant -> bake offsets into the pointers.
__device__ __forceinline__ void async_copy_b128(const float* g, float* l) {
#if __has_builtin(__builtin_amdgcn_global_load_async_to_lds_b128)
  __builtin_amdgcn_global_load_async_to_lds_b128((gv4i_p)g, (sv4i_p)l, 0, 0);
#else
  unsigned laddr = (unsigned)(size_t)(__attribute__((address_space(3))) float*)l;
  asm volatile("global_load_async_to_lds_b128 %0, %1, off"
               :: "v"(laddr), "v"(g) : "memory");
#endif
}

__device__ __forceinline__ void wait_async0() {
#if __has_builtin(__builtin_amdgcn_s_wait_asynccnt)
  __builtin_amdgcn_s_wait_asynccnt(0);
#else
  asm volatile("s_wait_asynccnt 0x0" ::: "memory");
#endif
}

// ---------------------------------------------------------------------------
// Kernel 1: gathered fp32 GEMM via V_WMMA_F32_16X16X4_F32 + masked max over P
// grid = (ceil(V/128), B), block = 256 threads (8 waves), each wave owns 16 cols
// A chunks double-buffered in LDS via async DMA, overlapped with WMMA compute.
// ---------------------------------------------------------------------------
__global__ __launch_bounds__(256) void sememe_gemm_max(
    const float* __restrict__ h,          // [B,S,H]
    const int*   __restrict__ index,      // [B,P]
    const int*   __restrict__ index_mask, // [B,P]
    const float* __restrict__ W,          // [V,H]
    const float* __restrict__ bias,       // [V]
    float*       __restrict__ score)      // [B,V]
{
  __shared__ float A_lds[2][PP * LDA];    // 2 x 36 KB gathered A chunks [256 x 32]
  __shared__ int   mask_lds[PP];
  __shared__ int   srow_lds[PP];

  const int b    = blockIdx.y;
  const int tid  = threadIdx.x;
  const int wave = tid >> 5;
  const int lane = tid & 31;
  const int lm   = lane & 15;   // lane within half
  const int hi   = lane >> 4;   // half: K offset 0/2, D rows r / r+8

  srow_lds[tid] = index[b * PP + tid];
  mask_lds[tid] = index_mask[b * PP + tid];

  const int n0    = blockIdx.x * 128 + wave * 16;
  const int vcol  = n0 + lm;
  const int vc    = vcol < VV ? vcol : (VV - 1);      // clamp tail, masked on store
  const float* __restrict__ wrow  = W + (size_t)vc * HH;
  const float* __restrict__ myrow = h + ((size_t)b * SS + srow_lds[tid]) * HH;

  v8f acc[16];
#pragma unroll
  for (int m = 0; m < 16; ++m)
    acc[m] = (v8f){0.f, 0.f, 0.f, 0.f, 0.f, 0.f, 0.f, 0.f};

  // issue DMA for chunk 0 (thread t owns piece-row t: 32 floats = 8 x b128)
#pragma unroll
  for (int j = 0; j < KC / 4; ++j)
    async_copy_b128(myrow + j * 4, &A_lds[0][tid * LDA + j * 4]);

  for (int c = 0; c < NCHUNK; ++c) {
    const int buf = c & 1;
    const int k0  = c * KC;

    wait_async0();        // our DMA writes for chunk c are in LDS
    __syncthreads();      // chunk c visible to all waves

    if (c + 1 < NCHUNK) { // overlap: DMA chunk c+1 while computing chunk c
#pragma unroll
      for (int j = 0; j < KC / 4; ++j)
        async_copy_b128(myrow + k0 + KC + j * 4,
                        &A_lds[buf ^ 1][tid * LDA + j * 4]);
    }

    // B fragments for this K chunk, reused across all 16 M tiles
    v2f bfrag[KC / 4];
#pragma unroll
    for (int ks = 0; ks < KC / 4; ++ks)
      bfrag[ks] = *(const v2f*)(wrow + k0 + ks * 4 + hi * 2);

#pragma unroll
    for (int m = 0; m < 16; ++m) {
#pragma unroll
      for (int ks = 0; ks < KC / 4; ++ks) {
        v2f af = *(const v2f*)(&A_lds[buf][(m * 16 + lm) * LDA + ks * 4 + hi * 2]);
        acc[m] = __builtin_amdgcn_wmma_f32_16x16x4_f32(
            false, af, false, bfrag[ks], (short)0, acc[m], false, false);
      }
    }
    __syncthreads();      // all waves done reading buf before it is overwritten
  }

  // --- epilogue: bias, mask -> -1e7 exactly (matches reference), max over P ---
  const float bval = bias[vc];
  float colmax = -1e7f;
#pragma unroll
  for (int m = 0; m < 16; ++m) {
#pragma unroll
    for (int r = 0; r < 8; ++r) {
      const int row = m * 16 + r + hi * 8;   // D layout: VGPR r -> rows r / r+8
      const float v = mask_lds[row] ? (acc[m][r] + bval) : -1e7f;
      colmax = fmaxf(colmax, v);
    }
  }
  colmax = fmaxf(colmax, __shfl_xor(colmax, 16, 32));  // merge the two halves
  if (hi == 0 && vcol < VV)
    score[(size_t)b * VV + vcol] = colmax;
}

// ---------------------------------------------------------------------------
// Kernel 2: stable descending rank-sort (== jnp.argsort(-score)) + loss
// one block per batch row
// ---------------------------------------------------------------------------
__global__ __launch_bounds__(256) void sememe_sort_loss(
    const float* __restrict__ score,  // [B,V]
    const float* __restrict__ y,      // [B,V]
    int*         __restrict__ out_idx,// [B,V]
    float*       __restrict__ loss)   // [1], pre-zeroed
{
  __shared__ float s_lds[VV];
  __shared__ float red[256];
  const int b   = blockIdx.x;
  const int tid = threadIdx.x;
  const float* srow = score + (size_t)b * VV;

  for (int j = tid; j < VV; j += 256) s_lds[j] = srow[j];
  __syncthreads();

  float acc = 0.f;
  for (int j = tid; j < VV; j += 256) {
    const float sj = s_lds[j];
    int rank = 0;
    for (int i = 0; i < VV; ++i) {
      const float si = s_lds[i];
      rank += (si > sj) || ((si == sj) && (i < j));
    }
    out_idx[(size_t)b * VV + rank] = j;

    const float t    = y[(size_t)b * VV + j];
    const float l1p  = log1pf(expf(-fabsf(sj)));
    const float ls_p = fminf(sj, 0.f) - l1p;    // log_sigmoid(s)
    const float ls_n = fminf(-sj, 0.f) - l1p;   // log_sigmoid(-s)
    acc += t * ls_p + (1.f - t) * ls_n;
  }

  red[tid] = acc;
  __syncthreads();
  for (int off = 128; off > 0; off >>= 1) {
    if (tid < off) red[tid] += red[tid + off];
    __syncthreads();
  }
  if (tid == 0) atomicAdd(loss, -red[0] / (float)(NB * VV));
}

// ---------------------------------------------------------------------------
extern "C" void kernel_launch(void* const* d_in, const int* in_sizes, int n_in,
                              void* d_out, int out_size, void* d_ws, size_t ws_size,
                              hipStream_t stream) {
  (void)in_sizes; (void)n_in; (void)d_ws; (void)ws_size; (void)out_size;
  const float* h          = (const float*)d_in[0];
  const float* y          = (const float*)d_in[1];
  const int*   index      = (const int*)d_in[2];
  const int*   index_mask = (const int*)d_in[3];
  const float* W          = (const float*)d_in[4];
  const float* bias       = (const float*)d_in[5];

  float* out   = (float*)d_out;
  float* loss  = out;                                    // [1]
  float* score = out + 1;                                // [B*V]
  int*   idxo  = (int*)(out + 1 + (size_t)NB * VV);      // [B*V] int32

  (void)hipMemsetAsync(loss, 0, sizeof(float), stream);  // graph-capturable

  dim3 g1((VV + 127) / 128, NB);
  sememe_gemm_max<<<g1, 256, 0, stream>>>(h, index, index_mask, W, bias, score);
  sememe_sort_loss<<<NB, 256, 0, stream>>>(score, y, idxo, loss);
}